// SelfSimilarity_48541720379771
// MI455X (gfx1250) — compile-verified
//
#include <hip/hip_runtime.h>

// ---------------- problem constants ----------------
constexpr int   kB      = 16;
constexpr int   kT      = 1024;
constexpr int   kC      = 128;
constexpr int   kUpperN = 523776;   // T*(T-1)/2
constexpr int   kTOut   = 2046;     // final conv length
constexpr int   kBlkOut = 6;        // final outputs per conv block (2046 = 6*341)
constexpr int   kBlkPB  = 341;      // conv blocks per batch
constexpr float kLeak   = 0.2f;

typedef __attribute__((ext_vector_type(2))) float v2f;
typedef __attribute__((ext_vector_type(8))) float v8f;

// ---------------- kernel 1: row norms xx[b,t] = sum_c x^2 ----------------
__global__ __launch_bounds__(256) void row_norms_kernel(const float* __restrict__ x,
                                                        float* __restrict__ xx) {
  int t = blockIdx.x * blockDim.x + threadIdx.x;     // 0 .. kB*kT-1 (exact grid)
  const float4* p = (const float4*)(x + (size_t)t * kC);
  float s = 0.f;
#pragma unroll
  for (int i = 0; i < kC / 4; ++i) {
    float4 v = p[i];
    s = fmaf(v.x, v.x, s); s = fmaf(v.y, v.y, s);
    s = fmaf(v.z, v.z, s); s = fmaf(v.w, v.w, s);
  }
  xx[t] = s;
}

// ---------------- kernel 2: G = X X^T via fp32 WMMA, write sqrt upper-tri ----------------
// One wave per (batch, tile-row ti). A strip (16 rows x 128 K) lives in registers
// (32 x v2f per lane per the 16x4 fp32 A layout), B streamed per 16-col tile.
__global__ __launch_bounds__(256) void dist_wmma_kernel(const float* __restrict__ x,
                                                        const float* __restrict__ xx,
                                                        float* __restrict__ upper) {
  int wid   = blockIdx.x * (blockDim.x >> 5) + (threadIdx.x >> 5);  // 0 .. kB*64-1
  int lane  = threadIdx.x & 31;
  int b     = wid >> 6;
  int ti    = wid & 63;
  int i0    = ti << 4;
  int row   = lane & 15;    // M index for A, N index for B/C/D
  int khalf = lane >> 4;    // which K pair

  const float* Xb  = x  + (size_t)b * kT * kC;
  const float* xxb = xx + b * kT;
  float*       Ub  = upper + (size_t)b * kUpperN;

  // Preload A strip: lane holds A[row][k4*4 + 2*khalf + {0,1}] for k4 = 0..31
  v2f areg[32];
#pragma unroll
  for (int k4 = 0; k4 < 32; ++k4)
    areg[k4] = *(const v2f*)(Xb + (size_t)(i0 + row) * kC + k4 * 4 + khalf * 2);

  float xxi[8];
#pragma unroll
  for (int r = 0; r < 8; ++r) xxi[r] = xxb[i0 + r + 8 * khalf];

  for (int tj = ti; tj < 64; ++tj) {
    int j0 = tj << 4;
    v8f acc = {};
#pragma unroll
    for (int k4 = 0; k4 < 32; ++k4) {
      v2f breg = *(const v2f*)(Xb + (size_t)(j0 + row) * kC + k4 * 4 + khalf * 2);
      acc = __builtin_amdgcn_wmma_f32_16x16x4_f32(false, areg[k4], false, breg,
                                                  (short)0, acc, false, false);
    }
    float xxj = xxb[j0 + row];
#pragma unroll
    for (int r = 0; r < 8; ++r) {
      int i = i0 + r + 8 * khalf;   // global row (M) of this acc element
      int j = j0 + row;             // global col (N)
      if (j > i) {
        float d2 = fmaxf(xxi[r] + xxj - 2.0f * acc[r], 0.0f);
        unsigned u = (unsigned)i * (2u * kT - (unsigned)i - 1u) / 2u + (unsigned)(j - i - 1);
        Ub[u] = sqrtf(d2);
      }
    }
  }
}

// ---------------- kernel 3: fused 8-layer binary-tree conv ----------------
__device__ __forceinline__ float lrelu(float v) { return v > 0.f ? v : kLeak * v; }

// one 8->8 conv level: out[p][co] = lrelu(b[co] + sum_ci w[co][ci][0]*in[2p][ci] + w[co][ci][1]*in[2p+1][ci])
__device__ __forceinline__ void level8x8(const float* __restrict__ w,
                                         const float* __restrict__ bias,
                                         const float* __restrict__ in,
                                         float* __restrict__ outb,
                                         int nOut, int co, int pp) {
  float wr[16];
#pragma unroll
  for (int ci = 0; ci < 8; ++ci) {
    wr[2 * ci + 0] = w[(co * 8 + ci) * 2 + 0];
    wr[2 * ci + 1] = w[(co * 8 + ci) * 2 + 1];
  }
  float bz = bias[co];
  for (int p = pp; p < nOut; p += 32) {
    const float* L = in + 2 * p * 8;
    float acc = bz;
#pragma unroll
    for (int ci = 0; ci < 8; ++ci)
      acc = fmaf(wr[2 * ci], L[ci], fmaf(wr[2 * ci + 1], L[8 + ci], acc));
    outb[p * 8 + co] = lrelu(acc);
  }
}

__global__ __launch_bounds__(256) void conv_tree_kernel(
    const float* __restrict__ upper,
    const float* __restrict__ w0, const float* __restrict__ b0,
    const float* __restrict__ w1, const float* __restrict__ b1,
    const float* __restrict__ w2, const float* __restrict__ b2,
    const float* __restrict__ w3, const float* __restrict__ b3,
    const float* __restrict__ w4, const float* __restrict__ b4,
    const float* __restrict__ w5, const float* __restrict__ b5,
    const float* __restrict__ w6, const float* __restrict__ b6,
    const float* __restrict__ w7, const float* __restrict__ b7,
    float* __restrict__ out) {
  __shared__ float bufA[768 * 8];   // holds h1 / h3 / h5 / h7   (24 KB)
  __shared__ float bufB[384 * 8];   // holds h2 / h4 / h6        (12 KB)

  int blk = blockIdx.x;
  int b   = blk / kBlkPB;
  int t0  = (blk % kBlkPB) * kBlkOut;
  const float* src = upper + (size_t)b * kUpperN + (size_t)t0 * 256;

  int tid = threadIdx.x;
  int co  = tid & 7;
  int pp  = tid >> 3;               // 0..31 position group

  // Level 1: 1 -> 8 channels, 768 positions   (w0: [8][1][2])
  {
    float wa = w0[co * 2 + 0], wb = w0[co * 2 + 1], bz = b0[co];
    for (int p = pp; p < 768; p += 32) {
      float v = fmaf(wa, src[2 * p], fmaf(wb, src[2 * p + 1], bz));
      bufA[p * 8 + co] = lrelu(v);
    }
  }
  __syncthreads();
  level8x8(w1, b1, bufA, bufB, 384, co, pp); __syncthreads();
  level8x8(w2, b2, bufB, bufA, 192, co, pp); __syncthreads();
  level8x8(w3, b3, bufA, bufB,  96, co, pp); __syncthreads();
  level8x8(w4, b4, bufB, bufA,  48, co, pp); __syncthreads();
  level8x8(w5, b5, bufA, bufB,  24, co, pp); __syncthreads();
  level8x8(w6, b6, bufB, bufA,  12, co, pp); __syncthreads();

  // Level 8: 8 -> 1 channel, no activation    (w7: [1][8][2])
  if (tid < kBlkOut) {
    float acc = b7[0];
#pragma unroll
    for (int ci = 0; ci < 8; ++ci)
      acc = fmaf(w7[ci * 2 + 0], bufA[(2 * tid) * 8 + ci],
            fmaf(w7[ci * 2 + 1], bufA[(2 * tid + 1) * 8 + ci], acc));
    out[(size_t)b * kTOut + t0 + tid] = acc;
  }
}

// ---------------- launcher ----------------
extern "C" void kernel_launch(void* const* d_in, const int* in_sizes, int n_in,
                              void* d_out, int out_size, void* d_ws, size_t ws_size,
                              hipStream_t stream) {
  (void)in_sizes; (void)n_in; (void)out_size; (void)ws_size;
  const float* x = (const float*)d_in[0];
  const float* w[8];
  const float* bs[8];
  for (int i = 0; i < 8; ++i) {
    w[i]  = (const float*)d_in[1 + 2 * i];
    bs[i] = (const float*)d_in[2 + 2 * i];
  }
  float* xx    = (float*)d_ws;            // kB*kT floats (64 KB)
  float* upper = xx + (size_t)kB * kT;    // kB*kUpperN floats (~33.5 MB)
  float* out   = (float*)d_out;

  row_norms_kernel<<<(kB * kT) / 256, 256, 0, stream>>>(x, xx);
  dist_wmma_kernel<<<(kB * 64) / 8, 256, 0, stream>>>(x, xx, upper);
  conv_tree_kernel<<<kB * kBlkPB, 256, 0, stream>>>(upper,
      w[0], bs[0], w[1], bs[1], w[2], bs[2], w[3], bs[3],
      w[4], bs[4], w[5], bs[5], w[6], bs[6], w[7], bs[7], out);
}